// NTN_2911987826813
// MI455X (gfx1250) — compile-verified
//
#include <hip/hip_runtime.h>

typedef __attribute__((ext_vector_type(2))) float v2f;
typedef __attribute__((ext_vector_type(4))) float f32x4;
typedef __attribute__((ext_vector_type(8))) float v8f;

constexpr int B_ = 64, M_ = 4096, D_ = 128, O_ = 16;
constexpr int ROWS = 128;      // M-rows per workgroup (8 waves x 16)
constexpr int LPAD = D_ + 4;   // LDS row stride in floats: 132 -> bank-conflict free

// ---------------------------------------------------------------------------
// Kernel 1: Utilde[b,o,k] = sum_d x1[b,d]*W[o,d,k] + V[o,D+k]
//           c[b,o]       = sum_d x1[b,d]*V[o,d]   + bias[o]
// grid (O, B), block D threads. Tiny vs. main GEMM (33 MFLOP total).
// ---------------------------------------------------------------------------
__global__ void __launch_bounds__(D_)
ntn_prep(const float* __restrict__ x1, const float* __restrict__ W,
         const float* __restrict__ V, const float* __restrict__ bias,
         float* __restrict__ uT, float* __restrict__ cArr) {
  const int o = blockIdx.x, b = blockIdx.y, k = threadIdx.x;
  __shared__ float sx[D_];
  __shared__ float red[D_];
  sx[k] = x1[b * D_ + k];
  __syncthreads();
  float acc = V[o * 2 * D_ + D_ + k];          // v2[o,k]  (V is (O,2D,1))
  const float* Wo = W + o * D_ * D_;
  #pragma unroll 8
  for (int d = 0; d < D_; ++d) acc = fmaf(sx[d], Wo[d * D_ + k], acc);
  uT[((size_t)b * O_ + o) * D_ + k] = acc;

  red[k] = sx[k] * V[o * 2 * D_ + k];          // x1[b,k]*v1[o,k]
  __syncthreads();
  for (int s = D_ / 2; s > 0; s >>= 1) {
    if (k < s) red[k] += red[k + s];
    __syncthreads();
  }
  if (k == 0) cArr[b * O_ + o] = red[0] + bias[o];
}

// ---------------------------------------------------------------------------
// Kernel 2: out[b,m,o] = relu( x2[b,m,:] . Utilde[b,o,:] + c[b,o] )
// One wave -> one 16x16 (MxO) tile via V_WMMA_F32_16X16X4_F32, K swept by 4.
// grid (M/ROWS, B), block 256 (8 waves).
// ---------------------------------------------------------------------------
__global__ void __launch_bounds__(256)
ntn_main(const float* __restrict__ x2, const float* __restrict__ uT,
         const float* __restrict__ cArr, float* __restrict__ out) {
  __shared__ float sX[ROWS * LPAD];   // 128 x 132 floats = 66 KiB
  __shared__ float sU[O_ * LPAD];     //  16 x 132 floats ~ 8 KiB

  const int b   = blockIdx.y;
  const int m0  = blockIdx.x * ROWS;
  const int tid = threadIdx.x;

  // ---- stage x2 tile: coalesced b128 global loads -> padded LDS ----
  const float* gX = x2 + ((size_t)b * M_ + m0) * D_;
  #pragma unroll
  for (int i = 0; i < (ROWS * D_ / 4) / 256; ++i) {   // 16 float4 per thread
    int idx = i * 256 + tid;
    int r = idx >> 5;       // D/4 = 32 float4 per row
    int c4 = idx & 31;
    f32x4 v = *(const f32x4*)(gX + r * D_ + c4 * 4);
    *(f32x4*)(&sX[r * LPAD + c4 * 4]) = v;            // 528B row stride, 16B aligned
  }
  // ---- stage Utilde[b] (16x128) ----
  const float* gU = uT + (size_t)b * O_ * D_;
  #pragma unroll
  for (int i = 0; i < (O_ * D_ / 4) / 256; ++i) {     // 2 float4 per thread
    int idx = i * 256 + tid;
    int r = idx >> 5, c4 = idx & 31;
    f32x4 v = *(const f32x4*)(gU + r * D_ + c4 * 4);
    *(f32x4*)(&sU[r * LPAD + c4 * 4]) = v;
  }
  __syncthreads();

  const int lane = tid & 31;
  const int wave = tid >> 5;
  const int n    = lane & 15;          // O column (and A-row index within tile)
  const int kh   = (lane >> 4) * 2;    // per-half K sub-offset {0,2}

  // Accumulator pre-loaded with c[b,o]: identical across the 8 row-VGPRs.
  float cv = cArr[b * O_ + n];
  v8f acc = {cv, cv, cv, cv, cv, cv, cv, cv};

  // A lane layout (16x4 f32): row = lane&15, VGPR0/1 = K kh+0 / kh+1
  // B lane layout (4x16 f32): col = lane&15, VGPR0/1 = K kh+0 / kh+1
  const float* aRow = &sX[(wave * 16 + n) * LPAD + kh];
  const float* bRow = &sU[n * LPAD + kh];
  #pragma unroll
  for (int k0 = 0; k0 < D_; k0 += 4) {
    v2f a  = *(const v2f*)(aRow + k0);   // ds_load_b64, conflict-free
    v2f bb = *(const v2f*)(bRow + k0);
    acc = __builtin_amdgcn_wmma_f32_16x16x4_f32(
        /*neg_a=*/false, a, /*neg_b=*/false, bb,
        /*c_mod=*/(short)0, acc, /*reuse_a=*/false, /*reuse_b=*/false);
  }

  // C/D layout: VGPR v holds row (v + 8*(lane>>4)), col = lane&15
  float* gOut = out + ((size_t)b * M_ + m0 + wave * 16 + (lane >> 4) * 8) * O_ + n;
  #pragma unroll
  for (int v = 0; v < 8; ++v) {
    gOut[(size_t)v * O_] = fmaxf(acc[v], 0.0f);
  }
}

// ---------------------------------------------------------------------------
extern "C" void kernel_launch(void* const* d_in, const int* in_sizes, int n_in,
                              void* d_out, int out_size, void* d_ws, size_t ws_size,
                              hipStream_t stream) {
  const float* x1   = (const float*)d_in[0];
  const float* x2   = (const float*)d_in[1];
  const float* W    = (const float*)d_in[2];
  const float* V    = (const float*)d_in[3];
  const float* bias = (const float*)d_in[4];

  float* uT = (float*)d_ws;                         // B*O*D floats (512 KiB)
  float* cA = uT + (size_t)B_ * O_ * D_;            // B*O floats

  ntn_prep<<<dim3(O_, B_), D_, 0, stream>>>(x1, W, V, bias, uT, cA);
  ntn_main<<<dim3(M_ / ROWS, B_), 256, 0, stream>>>(x2, uT, cA, (float*)d_out);
}